// GibbsEncoder_20461224198819
// MI455X (gfx1250) — compile-verified
//
#include <hip/hip_runtime.h>
#include <math.h>

typedef __attribute__((ext_vector_type(2))) float v2f;
typedef __attribute__((ext_vector_type(8))) float v8f;

#define G      1000
#define H      128
#define LATENT 32
#define BATCH  512

// ---------------------------------------------------------------------------
// Stage 0a: build column mask (mask[g] = 1 if g appears in masked_genes).
// One block; init then scatter with a barrier between (non-atomic store of 1
// is fine: all writers write the same value).
// ---------------------------------------------------------------------------
__global__ void __launch_bounds__(1024) ge_mask_kernel(const int* __restrict__ mg,
                                                       int* __restrict__ mask) {
    int t = threadIdx.x;
    if (t < G) mask[t] = 0;
    __syncthreads();
    if (t < BATCH) mask[mg[t]] = 1;
}

// ---------------------------------------------------------------------------
// Stage 0b: xlog[b,g] = mask[g] ? 0 : log1p(x[b,g])
// ---------------------------------------------------------------------------
__global__ void __launch_bounds__(256) ge_log_kernel(const float* __restrict__ x,
                                                     const int* __restrict__ mask,
                                                     float* __restrict__ xlog) {
    int i = blockIdx.x * 256 + threadIdx.x;
    if (i >= BATCH * G) return;
    int g = i % G;
    xlog[i] = mask[g] ? 0.0f : log1pf(x[i]);
}

// ---------------------------------------------------------------------------
// Stage 1 (memory-bound, ~262 MB streamed): per-sample gathered matvec.
// h0[b,h] = sum_g amats_table[mg[b]][h*G + g] * xlog[b,g] + bvecs[mg[b],h]
// One block per sample. 8 waves; wave w owns rows h = w*16 .. w*16+15,
// processed in pairs for memory-level parallelism. Each row iteration issues
// a full-wave contiguous 512B float4 load (perfectly coalesced).
// ---------------------------------------------------------------------------
__global__ void __launch_bounds__(256) ge_matvec_kernel(const float* __restrict__ xlog,
                                                        const int* __restrict__ mg,
                                                        const float* __restrict__ amats,
                                                        const float* __restrict__ bvecs,
                                                        float* __restrict__ h0) {
    __shared__ float xs[1024];
    int b = blockIdx.x;
    int t = threadIdx.x;
    for (int i = t; i < 1024; i += 256) xs[i] = (i < G) ? xlog[b * G + i] : 0.0f;
    __syncthreads();

    int gene = mg[b];
    const float* arow = amats + (size_t)gene * (size_t)(G * H);
    int wave = t >> 5, lane = t & 31;
    int g0 = lane * 4;

    for (int r = 0; r < 16; r += 2) {
        int h = wave * 16 + r;
        const float* ap0 = arow + (size_t)h * G;
        const float* ap1 = ap0 + G;
        // hint the next row-pair into cache while we chew this one
        __builtin_prefetch(ap0 + 2 * G + g0, 0, 0);
        float acc0 = 0.0f, acc1 = 0.0f;
#pragma unroll
        for (int it = 0; it < 8; ++it) {
            int g = it * 128 + g0;
            if (g < G) {
                float4 xv = *(const float4*)(xs + g);
                float4 a0 = *(const float4*)(ap0 + g);
                float4 a1 = *(const float4*)(ap1 + g);
                acc0 = fmaf(a0.x, xv.x, acc0); acc0 = fmaf(a0.y, xv.y, acc0);
                acc0 = fmaf(a0.z, xv.z, acc0); acc0 = fmaf(a0.w, xv.w, acc0);
                acc1 = fmaf(a1.x, xv.x, acc1); acc1 = fmaf(a1.y, xv.y, acc1);
                acc1 = fmaf(a1.z, xv.z, acc1); acc1 = fmaf(a1.w, xv.w, acc1);
            }
        }
#pragma unroll
        for (int off = 16; off > 0; off >>= 1) {
            acc0 += __shfl_xor(acc0, off, 32);
            acc1 += __shfl_xor(acc1, off, 32);
        }
        if (lane == 0) {
            h0[b * H + h]     = acc0 + bvecs[gene * H + h];
            h0[b * H + h + 1] = acc1 + bvecs[gene * H + h + 1];
        }
    }
}

// ---------------------------------------------------------------------------
// Stage 2: fused head using V_WMMA_F32_16X16X4_F32 (exact fp32 matrix path).
//   h1 = h0 @ w1 + b1 ; LayerNorm ; ReLU ; mean = h@w3+b3 ; scale = exp(h@w4+b4)
// One block per 16 rows of the batch (32 blocks, 256 threads = 8 waves).
//
// Fragment layouts (ISA 7.12.2, wave32):
//   A (16x4 f32): lane L holds rows M = L&15; VGPR j is K = 2*(L>>4)+j
//   B (4x16 f32): mirrored: VGPR j, lane-half h -> K = 2h+j, N = L&15
//   C/D (16x16 f32): VGPR r -> M = r + 8*(L>>4), N = L&15
// ---------------------------------------------------------------------------
__global__ void __launch_bounds__(256) ge_head_kernel(const float* __restrict__ h0,
                                                      const float* __restrict__ w1,
                                                      const float* __restrict__ b1,
                                                      const float* __restrict__ ln_scale,
                                                      const float* __restrict__ ln_bias,
                                                      const float* __restrict__ w3,
                                                      const float* __restrict__ b3,
                                                      const float* __restrict__ w4,
                                                      const float* __restrict__ b4,
                                                      float* __restrict__ out_mean,
                                                      float* __restrict__ out_scale) {
    __shared__ float hs[16 * 132];   // h0 tile (stride 132 avoids bank conflicts)
    __shared__ float hr[16 * 132];   // h1 / relu'd tile
    __shared__ float redS[256];
    __shared__ float redQ[256];
    __shared__ float muA[16];
    __shared__ float rsA[16];

    int t = threadIdx.x;
    int rowBase = blockIdx.x * 16;

    for (int idx = t; idx < 16 * H; idx += 256) {
        int r = idx >> 7, c = idx & 127;
        hs[r * 132 + c] = h0[(rowBase + r) * H + c];
    }
    __syncthreads();

    int wave = t >> 5, lane = t & 31;
    int half = lane >> 4, n = lane & 15;
    int m = lane & 15;           // A-fragment row index (same for both halves)

    // ---- GEMM1: wave `wave` computes columns [wave*16, wave*16+16) of h@w1
    {
        int cbase = wave * 16 + n;
        v8f acc = {};
        for (int k = 0; k < H; k += 4) {
            v2f av, bv;
            av.x = hs[m * 132 + k + 2 * half];
            av.y = hs[m * 132 + k + 2 * half + 1];
            bv.x = w1[(k + 2 * half) * H + cbase];
            bv.y = w1[(k + 2 * half + 1) * H + cbase];
            acc = __builtin_amdgcn_wmma_f32_16x16x4_f32(false, av, false, bv,
                                                        (short)0, acc, false, false);
        }
        float bias = b1[cbase];
#pragma unroll
        for (int r = 0; r < 8; ++r)
            hr[(r + 8 * half) * 132 + cbase] = acc[r] + bias;
    }
    __syncthreads();

    // ---- LayerNorm + ReLU over rows of 128 (16 threads per row)
    {
        int r = t >> 4, cs = t & 15;
        float s = 0.0f, q = 0.0f;
        for (int c = cs; c < H; c += 16) { float v = hr[r * 132 + c]; s += v; q += v * v; }
        redS[t] = s; redQ[t] = q;
    }
    __syncthreads();
    if (t < 16) {
        float s = 0.0f, q = 0.0f;
        for (int i = 0; i < 16; ++i) { s += redS[t * 16 + i]; q += redQ[t * 16 + i]; }
        float mu  = s * (1.0f / H);
        float var = q * (1.0f / H) - mu * mu;
        muA[t] = mu;
        rsA[t] = rsqrtf(var + 1e-6f);
    }
    __syncthreads();
    {
        int r = t >> 4, cs = t & 15;
        float mu = muA[r], rs = rsA[r];
        for (int c = cs; c < H; c += 16) {
            float v = (hr[r * 132 + c] - mu) * rs * ln_scale[c] + ln_bias[c];
            hr[r * 132 + c] = v > 0.0f ? v : 0.0f;
        }
    }
    __syncthreads();

    // ---- GEMM2: waves 0,1 -> mean tiles; waves 2,3 -> log_var tiles (exp'd)
    if (wave < 4) {
        const float* W  = (wave < 2) ? w3 : w4;
        const float* Bv = (wave < 2) ? b3 : b4;
        int ct = wave & 1;
        int cbase = ct * 16 + n;
        v8f acc = {};
        for (int k = 0; k < H; k += 4) {
            v2f av, bv;
            av.x = hr[m * 132 + k + 2 * half];
            av.y = hr[m * 132 + k + 2 * half + 1];
            bv.x = W[(k + 2 * half) * LATENT + cbase];
            bv.y = W[(k + 2 * half + 1) * LATENT + cbase];
            acc = __builtin_amdgcn_wmma_f32_16x16x4_f32(false, av, false, bv,
                                                        (short)0, acc, false, false);
        }
        float bias = Bv[cbase];
        if (wave < 2) {
#pragma unroll
            for (int r = 0; r < 8; ++r)
                out_mean[(rowBase + r + 8 * half) * LATENT + cbase] = acc[r] + bias;
        } else {
#pragma unroll
            for (int r = 0; r < 8; ++r)
                out_scale[(rowBase + r + 8 * half) * LATENT + cbase] = expf(acc[r] + bias);
        }
    }
}

// ---------------------------------------------------------------------------
extern "C" void kernel_launch(void* const* d_in, const int* in_sizes, int n_in,
                              void* d_out, int out_size, void* d_ws, size_t ws_size,
                              hipStream_t stream) {
    const float* x     = (const float*)d_in[0];
    const int*   mg    = (const int*)  d_in[1];
    const float* amats = (const float*)d_in[2];
    const float* bvecs = (const float*)d_in[3];
    const float* w1    = (const float*)d_in[4];
    const float* b1    = (const float*)d_in[5];
    const float* lns   = (const float*)d_in[6];
    const float* lnb   = (const float*)d_in[7];
    const float* w3    = (const float*)d_in[8];
    const float* b3    = (const float*)d_in[9];
    const float* w4    = (const float*)d_in[10];
    const float* b4    = (const float*)d_in[11];

    char*  ws    = (char*)d_ws;
    int*   mask  = (int*)ws;                                   // 4 KB slot
    float* xlog  = (float*)(ws + 4096);                        // 512*1000 f32
    float* h0    = (float*)(ws + 4096 + (size_t)BATCH * G * sizeof(float)); // 512*128 f32

    float* out_mean  = (float*)d_out;                          // (512,32)
    float* out_scale = out_mean + BATCH * LATENT;              // (512,32)

    ge_mask_kernel<<<1, 1024, 0, stream>>>(mg, mask);
    ge_log_kernel<<<(BATCH * G + 255) / 256, 256, 0, stream>>>(x, mask, xlog);
    ge_matvec_kernel<<<BATCH, 256, 0, stream>>>(xlog, mg, amats, bvecs, h0);
    ge_head_kernel<<<BATCH / 16, 256, 0, stream>>>(h0, w1, b1, lns, lnb,
                                                   w3, b3, w4, b4, out_mean, out_scale);
}